// SelfSSM_360777252919
// MI455X (gfx1250) — compile-verified
//
#include <hip/hip_runtime.h>
#include <math.h>

#define B_  4
#define L_  1024
#define D_  1024
#define R_  64
#define N_  16
#define RC_ 96            // R_ + 2*N_
#define ROWS_ (B_ * L_)   // 4096

#define CH_ 8             // chunks along L
#define CL_ (L_ / CH_)    // 128 steps per chunk
#define TS_ 16            // timesteps of B/C staged per LDS fill

typedef float v2f __attribute__((ext_vector_type(2)));
typedef float v8f __attribute__((ext_vector_type(8)));

// ---------------------------------------------------------------------------
// GEMM1: dbc[4096 x 96] = x[4096 x 1024] @ x_proj_w^T   (x_proj_w is [96,1024])
// One wave per 16x16 tile, K marched 4 at a time via v_wmma_f32_16x16x4_f32.
// ---------------------------------------------------------------------------
__global__ void xproj_wmma_kernel(const float* __restrict__ x,
                                  const float* __restrict__ w,   // (96, 1024)
                                  float* __restrict__ dbc) {
  const int wave = (blockIdx.x * blockDim.x + threadIdx.x) >> 5;
  const int lane = threadIdx.x & 31;
  const int NT = RC_ / 16;                 // 6 col tiles
  const int mt = wave / NT;
  const int nt = wave % NT;
  const int row0 = mt * 16, col0 = nt * 16;
  const int half = lane >> 4, lid = lane & 15;

  v8f acc = {};
  const float* arow = x + (size_t)(row0 + lid) * D_ + 2 * half;
  const float* brow = w + (size_t)(col0 + lid) * D_ + 2 * half;
  for (int k = 0; k < D_; k += 4) {
    v2f a = *(const v2f*)(arow + k);
    v2f b = *(const v2f*)(brow + k);
    acc = __builtin_amdgcn_wmma_f32_16x16x4_f32(false, a, false, b,
                                                (short)0, acc, false, false);
  }
  float* drow = dbc + (size_t)(row0 + 8 * half) * RC_ + col0 + lid;
#pragma unroll
  for (int v = 0; v < 8; ++v) drow[(size_t)v * RC_] = acc[v];
}

// ---------------------------------------------------------------------------
// GEMM2: delta[4096 x 1024] = softplus(delta_r[4096 x 64] @ dt_proj_w^T + bias)
// delta_r = dbc[:, 0:64] (row stride 96). dt_proj_w is [1024, 64].
// ---------------------------------------------------------------------------
__global__ void dtproj_wmma_kernel(const float* __restrict__ dbc,
                                   const float* __restrict__ wdt,  // (1024, 64)
                                   const float* __restrict__ bias, // (1024)
                                   float* __restrict__ delta) {
  const int wave = (blockIdx.x * blockDim.x + threadIdx.x) >> 5;
  const int lane = threadIdx.x & 31;
  const int NT = D_ / 16;                  // 64 col tiles
  const int mt = wave / NT;
  const int nt = wave % NT;
  const int row0 = mt * 16, col0 = nt * 16;
  const int half = lane >> 4, lid = lane & 15;

  v8f acc = {};
  const float* arow = dbc + (size_t)(row0 + lid) * RC_ + 2 * half;  // cols 0..63
  const float* brow = wdt + (size_t)(col0 + lid) * R_ + 2 * half;
#pragma unroll
  for (int k = 0; k < R_; k += 4) {
    v2f a = *(const v2f*)(arow + k);
    v2f b = *(const v2f*)(brow + k);
    acc = __builtin_amdgcn_wmma_f32_16x16x4_f32(false, a, false, b,
                                                (short)0, acc, false, false);
  }
  const float bz = bias[col0 + lid];
  float* drow = delta + (size_t)(row0 + 8 * half) * D_ + col0 + lid;
#pragma unroll
  for (int v = 0; v < 8; ++v) {
    float z = acc[v] + bz;
    float sp = (z > 20.0f) ? z : log1pf(__expf(z));   // softplus
    drow[(size_t)v * D_] = sp;
  }
}

// ---------------------------------------------------------------------------
// Chunked scan, pass A (EMIT=0) and pass C (EMIT=1).
// Grid: (B_ * CH_ * D_/256) blocks of 256; block owns one (b, chunk, d-group).
// Pass A: local scan with h0=0; writes hloc + closed-form decay exp(a*sum(du)).
// Pass C: local scan seeded with hstart; writes y.
// B/C rows for TS_ timesteps staged per LDS fill (2 barriers / 16 steps).
// ---------------------------------------------------------------------------
template <bool EMIT>
__global__ void scan_chunk_kernel(const float* __restrict__ x,
                                  const float* __restrict__ dbc,
                                  const float* __restrict__ delta,
                                  const float* __restrict__ A_log,
                                  const float* __restrict__ Dp,
                                  const float* __restrict__ hstart,
                                  float* __restrict__ hloc,
                                  float* __restrict__ decay,
                                  float* __restrict__ y) {
  __shared__ float sBC[TS_][2 * N_];
  const int DG = D_ / 256;                       // 4 d-groups
  const int b  = blockIdx.x / (CH_ * DG);
  const int rm = blockIdx.x % (CH_ * DG);
  const int c  = rm / DG;
  const int d  = (rm % DG) * 256 + threadIdx.x;
  const int l0 = c * CL_;
  const size_t base = (size_t)b * L_;
  const size_t sidx0 = ((size_t)(b * CH_ + c) * N_) * D_ + d;  // [b][c][n][d]

  float a[N_], h[N_];
#pragma unroll
  for (int n = 0; n < N_; ++n) {
    a[n] = -__expf(A_log[(size_t)d * N_ + n]);
    h[n] = EMIT ? hstart[sidx0 + (size_t)n * D_] : 0.0f;
  }
  const float dpv = Dp[d];
  float sumdu = 0.0f;

  for (int s = 0; s < CL_ / TS_; ++s) {
    __syncthreads();
#pragma unroll
    for (int j = 0; j < (TS_ * 2 * N_) / 256; ++j) {   // 512 floats, 2/thread
      const int idx = j * 256 + threadIdx.x;
      const int st = idx >> 5, cc = idx & 31;
      sBC[st][cc] = dbc[(base + l0 + s * TS_ + st) * RC_ + R_ + cc];
    }
    __syncthreads();

    if (s + 1 < CL_ / TS_) {                           // gfx1250 global_prefetch_b8
      __builtin_prefetch(&delta[(base + l0 + (s + 1) * TS_) * D_ + d], 0, 0);
      __builtin_prefetch(&x[(base + l0 + (s + 1) * TS_) * D_ + d], 0, 0);
    }

    for (int t = 0; t < TS_; ++t) {
      const size_t li = (base + l0 + s * TS_ + t) * D_ + d;
      const float du = delta[li];
      const float xv = x[li];
      const float dx = du * xv;
      if (!EMIT) sumdu += du;
      float yacc = 0.0f;
#pragma unroll
      for (int n = 0; n < N_; ++n) {
        const float dA = __expf(du * a[n]);
        h[n] = dA * h[n] + dx * sBC[t][n];             // h = exp(dt*A)*h + (dt*x)*B
        if (EMIT) yacc += h[n] * sBC[t][N_ + n];       // y += h*C
      }
      if (EMIT) y[li] = yacc + dpv * xv;
    }
  }

  if (!EMIT) {
#pragma unroll
    for (int n = 0; n < N_; ++n) {
      hloc[sidx0 + (size_t)n * D_]  = h[n];
      decay[sidx0 + (size_t)n * D_] = __expf(a[n] * sumdu);  // prod exp = exp(a*sum)
    }
  }
}

// ---------------------------------------------------------------------------
// Pass B: per (b,d), sequential combine over CH_ chunks (tiny):
//   hstart[0] = 0;  hstart[c] = decay[c-1]*hstart[c-1] + hloc[c-1]
// ---------------------------------------------------------------------------
__global__ void combine_kernel(const float* __restrict__ hloc,
                               const float* __restrict__ decay,
                               float* __restrict__ hstart) {
  const int t = blockIdx.x * blockDim.x + threadIdx.x;   // 0..4095
  const int b = t / D_;
  const int d = t % D_;
  float h[N_];
#pragma unroll
  for (int n = 0; n < N_; ++n) h[n] = 0.0f;
  for (int c = 0; c < CH_; ++c) {
    const size_t idx0 = ((size_t)(b * CH_ + c) * N_) * D_ + d;
#pragma unroll
    for (int n = 0; n < N_; ++n) {
      const size_t i = idx0 + (size_t)n * D_;
      hstart[i] = h[n];
      h[n] = decay[i] * h[n] + hloc[i];
    }
  }
}

// ---------------------------------------------------------------------------
// LayerNorm (ddof=1) over D per (b,l) row:  out = ga*(y-mean)/(std+eps)+gb
// ---------------------------------------------------------------------------
__global__ void ln_kernel(const float* __restrict__ y,
                          const float* __restrict__ ga,
                          const float* __restrict__ gb,
                          float* __restrict__ out) {
  __shared__ float s1[256], s2[256];
  const int row = blockIdx.x;
  const float* yr = y + (size_t)row * D_;

  float sum = 0.0f, sq = 0.0f;
  for (int i = threadIdx.x; i < D_; i += 256) {
    const float v = yr[i];
    sum += v;
    sq += v * v;
  }
  s1[threadIdx.x] = sum;
  s2[threadIdx.x] = sq;
  __syncthreads();
  for (int off = 128; off > 0; off >>= 1) {
    if (threadIdx.x < off) {
      s1[threadIdx.x] += s1[threadIdx.x + off];
      s2[threadIdx.x] += s2[threadIdx.x + off];
    }
    __syncthreads();
  }
  const float mean = s1[0] * (1.0f / (float)D_);
  float var = (s2[0] - (float)D_ * mean * mean) * (1.0f / (float)(D_ - 1));
  var = var < 0.0f ? 0.0f : var;
  const float inv = 1.0f / (sqrtf(var) + 1e-12f);

  float* orow = out + (size_t)row * D_;
  for (int i = threadIdx.x; i < D_; i += 256) {
    orow[i] = ga[i] * (yr[i] - mean) * inv + gb[i];
  }
}

// ---------------------------------------------------------------------------
extern "C" void kernel_launch(void* const* d_in, const int* in_sizes, int n_in,
                              void* d_out, int out_size, void* d_ws, size_t ws_size,
                              hipStream_t stream) {
  const float* x        = (const float*)d_in[0];  // (B,L,D)
  const float* x_proj_w = (const float*)d_in[1];  // (96, D)
  const float* dt_w     = (const float*)d_in[2];  // (D, R)
  const float* dt_b     = (const float*)d_in[3];  // (D,)
  const float* A_log    = (const float*)d_in[4];  // (D, N)
  const float* Dp       = (const float*)d_in[5];  // (D,)
  const float* ln_a     = (const float*)d_in[6];  // (D,)
  const float* ln_b     = (const float*)d_in[7];  // (D,)
  float* out = (float*)d_out;                     // (B,L,D)

  // Workspace (36 MB total):
  //   dbc    @  0 MB (1.5 MB)
  //   delta  @  2 MB (16 MB)
  //   hstart @ 18 MB ( 2 MB)
  //   ytmp   @ 20 MB (16 MB)   -- hloc/decay overlap ytmp: consumed (pass B)
  //   hloc   @ 20 MB ( 2 MB)      before ytmp is produced (pass C).
  //   decay  @ 22 MB ( 2 MB)
  char* ws = (char*)d_ws;
  float* dbc    = (float*)(ws);
  float* delta  = (float*)(ws + (size_t)2  * 1024 * 1024);
  float* hstart = (float*)(ws + (size_t)18 * 1024 * 1024);
  float* ytmp   = (float*)(ws + (size_t)20 * 1024 * 1024);
  float* hloc   = (float*)(ws + (size_t)20 * 1024 * 1024);
  float* decay  = (float*)(ws + (size_t)22 * 1024 * 1024);

  // GEMM1: (4096/16)*(96/16) = 1536 waves -> 192 blocks x 256 thr
  xproj_wmma_kernel<<<192, 256, 0, stream>>>(x, x_proj_w, dbc);

  // GEMM2: (4096/16)*(1024/16) = 16384 waves -> 2048 blocks x 256 thr
  dtproj_wmma_kernel<<<2048, 256, 0, stream>>>(dbc, dt_w, dt_b, delta);

  // Pass A: local chunk scans (1024 waves)
  scan_chunk_kernel<false><<<B_ * CH_ * (D_ / 256), 256, 0, stream>>>(
      x, dbc, delta, A_log, Dp, nullptr, hloc, decay, ytmp);

  // Pass B: combine chunk states (tiny)
  combine_kernel<<<16, 256, 0, stream>>>(hloc, decay, hstart);

  // Pass C: seeded chunk scans, emit y (1024 waves)
  scan_chunk_kernel<true><<<B_ * CH_ * (D_ / 256), 256, 0, stream>>>(
      x, dbc, delta, A_log, Dp, hstart, hloc, decay, ytmp);

  // LayerNorm: one block per (b,l) row
  ln_kernel<<<ROWS_, 256, 0, stream>>>(ytmp, ln_a, ln_b, out);
}